// lossFunc_30812095381620
// MI455X (gfx1250) — compile-verified
//
#include <hip/hip_runtime.h>
#include <hip/hip_bf16.h>
#include <cstdint>

#define NUM_Q    4096
#define MAX_STEP 256
#define BATCH    64
#define ROW      (2 * NUM_Q)           // 8192 floats per [b,s] row of `batch`
#define ROWS     (BATCH * MAX_STEP)    // 16384 rows

// ---------------------------------------------------------------------------
// Phase 1: stream `batch` (512 MB) through LDS via gfx1250 async-to-LDS DMA.
// Each block reduces one [b,s] row of 8192 floats to {answered, correct, qid}.
// The one-hot value is exactly 1.0, so:
//   code = sum v * (k <  Q ? 3 : 1)  in {0, 1, 3}  (pad / incorrect / correct)
//   idx  = sum v * (k mod Q)                       (== qid, exact in f32)
// ---------------------------------------------------------------------------
__global__ __launch_bounds__(256) void kt_rowscan(const float* __restrict__ batch,
                                                  int* __restrict__ rowinfo) {
  __shared__ __align__(16) float lds[ROW];   // 32 KB, mirrors the global row
  const int t   = threadIdx.x;
  const int row = blockIdx.x;
  const float* base = batch + (size_t)row * ROW;

  // Low 32 bits of a generic pointer to LDS == workgroup-relative LDS byte addr.
  const unsigned lbase = (unsigned)(uintptr_t)&lds[0];

  // Issue 8 async B128 DMA copies per thread (8 x 4 KB per block = one row).
  // Thread t, chunk j covers floats [j*1024 + t*4, +4).
#pragma unroll
  for (int j = 0; j < 8; ++j) {
    const unsigned byteoff = (unsigned)(j * 4096 + t * 16);
    const float*   g       = base + (size_t)(j * 1024 + t * 4);
    const unsigned l       = lbase + byteoff;
    asm volatile("global_load_async_to_lds_b128 %0, %1, off"
                 :: "v"(l), "v"(g)
                 : "memory");
  }

  float acc_code = 0.0f;
  float acc_idx  = 0.0f;

  // Async loads complete in order: after waiting to <=4 outstanding, chunks
  // 0..3 have landed; overlap their reduction with the tail of the DMA.
  asm volatile("s_wait_asynccnt 4" ::: "memory");
#pragma unroll
  for (int j = 0; j < 4; ++j) {                    // first half: k < Q, weight 3
    const int kbase = j * 1024 + t * 4;
    const float4 v = *reinterpret_cast<const float4*>(&lds[kbase]);
    const float km = (float)kbase;                 // kbase < 4096 -> k mod Q = k
    acc_code += 3.0f * ((v.x + v.y) + (v.z + v.w));
    acc_idx  += v.x * km + v.y * (km + 1.0f) + v.z * (km + 2.0f) + v.w * (km + 3.0f);
  }
  asm volatile("s_wait_asynccnt 0" ::: "memory");
#pragma unroll
  for (int j = 4; j < 8; ++j) {                    // second half: k >= Q, weight 1
    const int kbase = j * 1024 + t * 4;
    const float4 v = *reinterpret_cast<const float4*>(&lds[kbase]);
    const float km = (float)(kbase & (NUM_Q - 1)); // k mod Q
    acc_code += (v.x + v.y) + (v.z + v.w);
    acc_idx  += v.x * km + v.y * (km + 1.0f) + v.z * (km + 2.0f) + v.w * (km + 3.0f);
  }

  // Block reduction; reuse the (now fully consumed) LDS row buffer.
  __syncthreads();
  lds[t]       = acc_code;
  lds[256 + t] = acc_idx;
  __syncthreads();
  for (int off = 128; off > 0; off >>= 1) {
    if (t < off) {
      lds[t]       += lds[t + off];
      lds[256 + t] += lds[256 + t + off];
    }
    __syncthreads();
  }
  if (t == 0) {
    const float code = lds[0];
    const float idx  = lds[256];
    const int qid = (int)(idx + 0.5f);
    const int pk  = (qid << 2) | ((code > 2.5f) ? 2 : 0) | ((code > 0.5f) ? 1 : 0);
    rowinfo[row] = pk;
  }
}

// ---------------------------------------------------------------------------
// Phase 2: one wave32 per student. Gathers 255 scalars from `pred`
// (instead of streaming its full 256 MB), computes trimmed BCE + tail,
// and writes a per-student partial (no atomics -> bitwise deterministic).
// ---------------------------------------------------------------------------
__device__ __forceinline__ float clamped_log(float x)   { return fmaxf(logf(x),    -100.0f); }
__device__ __forceinline__ float clamped_log1m(float x) { return fmaxf(log1pf(-x), -100.0f); }

__global__ __launch_bounds__(32) void kt_loss(const float* __restrict__ pred,
                                              const float* __restrict__ target_q,
                                              const float* __restrict__ target_label,
                                              const int* __restrict__ rowinfo,
                                              float* __restrict__ partials) {
  const int b    = blockIdx.x;
  const int lane = threadIdx.x;
  const int* rows = rowinfo + b * MAX_STEP;

  // i = first s in [0, S-2] whose next row is answered (p > 0 <=> answered,
  // since pred in (0.01, 0.99)). If none answered, reference argmax gives 0.
  int mn = 0x7FFFFFFF;
  for (int s = lane; s < MAX_STEP - 1; s += 32)
    if (rows[s + 1] & 1) mn = min(mn, s);
  for (int off = 16; off > 0; off >>= 1)
    mn = min(mn, __shfl_xor(mn, off, 32));
  const int i = (mn == 0x7FFFFFFF) ? 0 : mn;

  // Trimmed BCE sum. Padded steps (p=0, a=0) contribute exactly 0 under the
  // reference's -100 log clamp, so skip them.
  float sum = 0.0f;
  for (int s = lane; s < MAX_STEP - 1; s += 32) {
    if (s < i) continue;
    const int pk = rows[s + 1];
    if (pk & 1) {
      const int   qid = pk >> 2;
      const float p   = pred[((size_t)b * MAX_STEP + s) * NUM_Q + qid];
      sum += (pk & 2) ? -clamped_log(p) : -clamped_log1m(p);
    }
  }
  for (int off = 16; off > 0; off >>= 1)
    sum += __shfl_xor(sum, off, 32);

  if (lane == 0) {
    const float q   = target_q[b];
    const float lab = target_label[b];
    const float tail = -(lab * clamped_log(q) + (1.0f - lab) * clamped_log1m(q));
    const float n = (float)((MAX_STEP - 1) - i + 1);
    partials[b] = (sum + tail) / n;
  }
}

// ---------------------------------------------------------------------------
// Phase 3: single wave32 reduces the 64 per-student partials in a fixed
// order and writes the scalar output.
// ---------------------------------------------------------------------------
__global__ __launch_bounds__(32) void kt_final(const float* __restrict__ partials,
                                               float* __restrict__ out) {
  const int lane = threadIdx.x;
  float v = partials[lane] + partials[lane + 32];
  for (int off = 16; off > 0; off >>= 1)
    v += __shfl_xor(v, off, 32);
  if (lane == 0) out[0] = v;
}

// ---------------------------------------------------------------------------
extern "C" void kernel_launch(void* const* d_in, const int* in_sizes, int n_in,
                              void* d_out, int out_size, void* d_ws, size_t ws_size,
                              hipStream_t stream) {
  (void)in_sizes; (void)n_in; (void)out_size; (void)ws_size;
  const float* pred         = (const float*)d_in[0];  // [64,256,4096] f32
  const float* target_q     = (const float*)d_in[1];  // [64,1]        f32
  const float* batch        = (const float*)d_in[2];  // [64,256,8192] f32
  const float* target_label = (const float*)d_in[3];  // [64]          f32
  float* out     = (float*)d_out;                     // scalar f32
  int*   rowinfo = (int*)d_ws;                        // 16384 ints = 64 KB
  float* partials = (float*)(rowinfo + ROWS);         // 64 floats after rowinfo

  kt_rowscan<<<ROWS, 256, 0, stream>>>(batch, rowinfo);
  kt_loss<<<BATCH, 32, 0, stream>>>(pred, target_q, target_label, rowinfo, partials);
  kt_final<<<1, 32, 0, stream>>>(partials, out);
}